// FraudDetectionGNN_59914793779321
// MI455X (gfx1250) — compile-verified
//
#include <hip/hip_runtime.h>

typedef __attribute__((ext_vector_type(2))) float v2f;
typedef __attribute__((ext_vector_type(8))) float v8f;

#define HID 64
#define F_IN 128

__device__ __forceinline__ v8f wmma_f32(v2f a, v2f b, v8f c) {
  // V_WMMA_F32_16X16X4_F32: D = A(16x4) * B(4x16) + C(16x16), full fp32
  return __builtin_amdgcn_wmma_f32_16x16x4_f32(false, a, false, b, (short)0, c,
                                               false, false);
}

__device__ __forceinline__ int imin(int a, int b) { return a < b ? a : b; }

__device__ __forceinline__ void lds_fence_wave() {
  // LDS pipe is in-order per wave; just stop compiler reordering and drain
  // DScnt so cross-lane data is in LDS before dependent reads issue.
  __builtin_amdgcn_wave_barrier();
  asm volatile("s_wait_dscnt 0x0" ::: "memory");
  __builtin_amdgcn_wave_barrier();
}

// ---------------------------------------------------------------- utilities
__global__ void fill_zero_kernel(float* __restrict__ p, int n) {
  int i = blockIdx.x * blockDim.x + threadIdx.x;
  if (i < n) p[i] = 0.0f;
}

__global__ void fill_zero4_kernel(float4* __restrict__ p, int n4) {
  int i = blockIdx.x * blockDim.x + threadIdx.x;
  if (i < n4) p[i] = make_float4(0.f, 0.f, 0.f, 0.f);
}

__global__ void deg_kernel(const int* __restrict__ dst, float* __restrict__ deg,
                           int E) {
  int e = blockIdx.x * blockDim.x + threadIdx.x;
  if (e < E) atomicAdd(&deg[dst[e]], 1.0f);
}

__global__ void dinv_kernel(float* __restrict__ deg_dinv, int n) {
  int i = blockIdx.x * blockDim.x + threadIdx.x;
  if (i < n) deg_dinv[i] = rsqrtf(deg_dinv[i] + 1.0f);  // +1 = self loop
}

// ------------------------------------------------ GCN dense GEMM (WMMA f32)
// hs[row, 0:64] = (A[row, :] @ W[:, 0:64]) * dinv[row]
// One wave per 16 rows; 4 N-tiles of 16; K stepped by 4.
// W staged in LDS as (k,k+1) pairs, pair-row stride 160 dwords (pad 32) so
// the two 16-lane halves read disjoint 32-bank windows: conflict-free b64.
template <int K>
__global__ __launch_bounds__(256) void gcn_gemm_scale_kernel(
    const float* __restrict__ A, const float* __restrict__ W,
    const float* __restrict__ dinv, float* __restrict__ hs, int M) {
  __shared__ float smW[(K / 2) * 160];
  v2f* __restrict__ smWv = (v2f*)smW;  // index: kp*80 + col

  // cooperative weight staging (before any exit; all threads hit the barrier)
  for (int p = threadIdx.x; p < (K / 2) * 64; p += 256) {
    const int kp = p >> 6, col = p & 63;
    v2f w;
    w.x = W[(size_t)(2 * kp) * HID + col];
    w.y = W[(size_t)(2 * kp + 1) * HID + col];
    smWv[kp * 80 + col] = w;
  }
  __syncthreads();

  const int lane = threadIdx.x & 31;
  const int wave = blockIdx.x * (blockDim.x >> 5) + (threadIdx.x >> 5);
  const int row0 = wave * 16;
  if (row0 >= M) return;  // wave-uniform: EXEC stays all-ones for WMMA

  const int m = lane & 15;          // A row (lanes 0-15 / 16-31 same rows)
  const int kh = (lane >> 4) << 1;  // K sub-pair: 0 for lo half, 2 for hi
  const float* arow = A + (size_t)imin(row0 + m, M - 1) * K;

  v8f c0 = {}, c1 = {}, c2 = {}, c3 = {};
  for (int k = 0; k < K; k += 4) {
    v2f a;
    a.x = arow[k + kh];
    a.y = arow[k + kh + 1];
    const v2f* wrow = smWv + ((k + kh) >> 1) * 80;
    c0 = wmma_f32(a, wrow[m], c0);
    c1 = wmma_f32(a, wrow[16 + m], c1);
    c2 = wmma_f32(a, wrow[32 + m], c2);
    c3 = wmma_f32(a, wrow[48 + m], c3);
  }

  // C layout: vgpr r, lanes 0-15 -> row r, lanes 16-31 -> row r+8; col = m+16n
  const int rbase = row0 + ((lane >> 4) << 3);
  float dv[8];
#pragma unroll
  for (int r = 0; r < 8; ++r) dv[r] = dinv[imin(rbase + r, M - 1)];
#pragma unroll
  for (int r = 0; r < 8; ++r) {
    if (rbase + r < M) {
      float* orow = hs + (size_t)(rbase + r) * HID + m;
      orow[0]  = c0[r] * dv[r];
      orow[16] = c1[r] * dv[r];
      orow[32] = c2[r] * dv[r];
      orow[48] = c3[r] * dv[r];
    }
  }
}

// ------------------------------------------------------- edge scatter (add)
// 16 threads per edge, float4 gather + 4 atomics each.
__global__ __launch_bounds__(256) void scatter_kernel(
    const float* __restrict__ hs, const int* __restrict__ src,
    const int* __restrict__ dst, float* __restrict__ acc, int E) {
  long long t = (long long)blockIdx.x * blockDim.x + threadIdx.x;
  int e = (int)(t >> 4);
  int f = (int)(t & 15) << 2;
  if (e >= E) return;
  const float4 v = *(const float4*)(hs + (size_t)src[e] * HID + f);
  float* a = acc + (size_t)dst[e] * HID + f;
  atomicAdd(a + 0, v.x);
  atomicAdd(a + 1, v.y);
  atomicAdd(a + 2, v.z);
  atomicAdd(a + 3, v.w);
}

// ----------------------------------------------- h = relu(dinv*(acc+hs)+b)
__global__ __launch_bounds__(256) void norm_relu_kernel(
    const float* __restrict__ hs, const float* __restrict__ dinv,
    const float* __restrict__ b, float* __restrict__ acc, int n) {
  long long t = (long long)blockIdx.x * blockDim.x + threadIdx.x;
  int i = (int)(t >> 4);
  int f = (int)(t & 15) << 2;
  if (i >= n) return;
  size_t idx = (size_t)i * HID + f;
  const float di = dinv[i];
  const float4 av = *(const float4*)(acc + idx);
  const float4 hv = *(const float4*)(hs + idx);
  const float4 bv = *(const float4*)(b + f);
  float4 o;
  o.x = fmaxf(di * (av.x + hv.x) + bv.x, 0.f);
  o.y = fmaxf(di * (av.y + hv.y) + bv.y, 0.f);
  o.z = fmaxf(di * (av.z + hv.z) + bv.z, 0.f);
  o.w = fmaxf(di * (av.w + hv.w) + bv.w, 0.f);
  *(float4*)(acc + idx) = o;  // in place; buffer becomes next layer input
}

// --------------------------------------------------------------- edge MLP
// One wave handles 16 edges: [16x144]@[144x64] -> relu -> [16x64]@[64x32]
// -> relu -> [16x32]@[32x2].  A-frags for layer 1 gathered straight from
// global (each element touched exactly once); Wm1/Wm2 staged in LDS in
// (k,k+1)-pair layout so every B fragment is one ds_load_b64.
__global__ __launch_bounds__(128) void edge_mlp_kernel(
    const float* __restrict__ h, const int* __restrict__ src,
    const int* __restrict__ dst, const float* __restrict__ eattr,
    const float* __restrict__ Wm1, const float* __restrict__ bm1,
    const float* __restrict__ Wm2, const float* __restrict__ bm2,
    const float* __restrict__ Wm3, const float* __restrict__ bm3,
    float* __restrict__ out, int E) {
  __shared__ float smW1[72 * 128];     // 144x64 as 72 pair-rows  (36.9 KB)
  __shared__ float smW2[32 * 64];      // 64x32  as 32 pair-rows  ( 8.2 KB)
  __shared__ float smScr[4][16 * 66];  // per-wave e1 (e2 overlays) (16.9 KB)
  v2f* __restrict__ smW1v = (v2f*)smW1;  // kp*64 + col
  v2f* __restrict__ smW2v = (v2f*)smW2;  // kp*32 + col

  for (int p = threadIdx.x; p < 72 * 64; p += 128) {
    const int kp = p >> 6, col = p & 63;
    v2f w;
    w.x = Wm1[(size_t)(2 * kp) * HID + col];
    w.y = Wm1[(size_t)(2 * kp + 1) * HID + col];
    smW1v[kp * 64 + col] = w;
  }
  for (int p = threadIdx.x; p < 32 * 32; p += 128) {
    const int kp = p >> 5, col = p & 31;
    v2f w;
    w.x = Wm2[(size_t)(2 * kp) * 32 + col];
    w.y = Wm2[(size_t)(2 * kp + 1) * 32 + col];
    smW2v[kp * 32 + col] = w;
  }
  __syncthreads();

  const int lane = threadIdx.x & 31;
  const int wv = threadIdx.x >> 5;
  const int e0 = (blockIdx.x * 4 + wv) * 16;
  if (e0 >= E) return;  // wave-uniform (after the workgroup barrier)

  const int m = lane & 15;
  const int kh = (lane >> 4) << 1;
  const int me = imin(e0 + m, E - 1);
  const float* hsrc = h + (size_t)src[me] * HID;
  const float* hdst = h + (size_t)dst[me] * HID;
  const float* ea = eattr + (size_t)me * 16;

  v8f c0 = {}, c1 = {}, c2 = {}, c3 = {};
  // K region 1: h[src] -> Wm1 rows 0..63
  for (int k = 0; k < 64; k += 4) {
    v2f a; a.x = hsrc[k + kh]; a.y = hsrc[k + kh + 1];
    const v2f* wrow = smW1v + ((k + kh) >> 1) * 64;
    c0 = wmma_f32(a, wrow[m], c0);
    c1 = wmma_f32(a, wrow[16 + m], c1);
    c2 = wmma_f32(a, wrow[32 + m], c2);
    c3 = wmma_f32(a, wrow[48 + m], c3);
  }
  // K region 2: h[dst] -> Wm1 rows 64..127
  for (int k = 0; k < 64; k += 4) {
    v2f a; a.x = hdst[k + kh]; a.y = hdst[k + kh + 1];
    const v2f* wrow = smW1v + ((64 + k + kh) >> 1) * 64;
    c0 = wmma_f32(a, wrow[m], c0);
    c1 = wmma_f32(a, wrow[16 + m], c1);
    c2 = wmma_f32(a, wrow[32 + m], c2);
    c3 = wmma_f32(a, wrow[48 + m], c3);
  }
  // K region 3: edge_attr -> Wm1 rows 128..143
  for (int k = 0; k < 16; k += 4) {
    v2f a; a.x = ea[k + kh]; a.y = ea[k + kh + 1];
    const v2f* wrow = smW1v + ((128 + k + kh) >> 1) * 64;
    c0 = wmma_f32(a, wrow[m], c0);
    c1 = wmma_f32(a, wrow[16 + m], c1);
    c2 = wmma_f32(a, wrow[32 + m], c2);
    c3 = wmma_f32(a, wrow[48 + m], c3);
  }

  // bias + relu, e1 -> LDS (C layout: row = r + 8*(lane>>4), col = m + 16n)
  float* sm1 = smScr[wv];
  {
    const float b0 = bm1[m], b1v = bm1[16 + m], b2v = bm1[32 + m],
                b3v = bm1[48 + m];
    const int rowh = (lane >> 4) << 3;
#pragma unroll
    for (int r = 0; r < 8; ++r) {
      float* p = sm1 + (rowh + r) * 66;
      p[m]      = fmaxf(c0[r] + b0, 0.0f);
      p[16 + m] = fmaxf(c1[r] + b1v, 0.0f);
      p[32 + m] = fmaxf(c2[r] + b2v, 0.0f);
      p[48 + m] = fmaxf(c3[r] + b3v, 0.0f);
    }
  }
  lds_fence_wave();

  // GEMM2: e1[16x64] @ Wm2[64x32], A-frags from LDS
  v8f d0 = {}, d1 = {};
  const float* arow2 = sm1 + m * 66;
  for (int k = 0; k < 64; k += 4) {
    v2f a; a.x = arow2[k + kh]; a.y = arow2[k + kh + 1];
    const v2f* wrow = smW2v + ((k + kh) >> 1) * 32;
    d0 = wmma_f32(a, wrow[m], d0);
    d1 = wmma_f32(a, wrow[16 + m], d1);
  }
  lds_fence_wave();  // e1 reads fully issued; safe to overlay e2 below

  // e2 overlays the (now dead) e1 scratch, stride 33
  {
    const float b0 = bm2[m], b1v = bm2[16 + m];
    const int rowh = (lane >> 4) << 3;
#pragma unroll
    for (int r = 0; r < 8; ++r) {
      float* p = sm1 + (rowh + r) * 33;
      p[m]      = fmaxf(d0[r] + b0, 0.0f);
      p[16 + m] = fmaxf(d1[r] + b1v, 0.0f);
    }
  }
  lds_fence_wave();

  // GEMM3: [16x32] @ Wm3[32x2]; lane -> (edge = lane>>1, class = lane&1)
  {
    const int eidx = lane >> 1, cls = lane & 1;
    const float* row = sm1 + eidx * 33;
    float accv = bm3[cls];
#pragma unroll
    for (int j = 0; j < 32; ++j) accv += row[j] * Wm3[j * 2 + cls];
    if (e0 + eidx < E) out[(size_t)(e0 + eidx) * 2 + cls] = accv;
  }
}

// --------------------------------------------------------------- launcher
extern "C" void kernel_launch(void* const* d_in, const int* in_sizes, int n_in,
                              void* d_out, int out_size, void* d_ws,
                              size_t ws_size, hipStream_t stream) {
  const float* x   = (const float*)d_in[0];
  const int*   ei  = (const int*)d_in[1];
  const float* ea  = (const float*)d_in[2];
  const float* W1  = (const float*)d_in[3];
  const float* b1  = (const float*)d_in[4];
  const float* W2  = (const float*)d_in[5];
  const float* b2  = (const float*)d_in[6];
  const float* W3  = (const float*)d_in[7];
  const float* b3  = (const float*)d_in[8];
  const float* Wm1 = (const float*)d_in[9];
  const float* bm1 = (const float*)d_in[10];
  const float* Wm2 = (const float*)d_in[11];
  const float* bm2 = (const float*)d_in[12];
  const float* Wm3 = (const float*)d_in[13];
  const float* bm3 = (const float*)d_in[14];
  float* out = (float*)d_out;

  const int N = in_sizes[0] / F_IN;
  const int E = in_sizes[1] / 2;
  const int* src = ei;
  const int* dst = ei + E;

  // workspace: dinv[N] | bufA[N*64] | bufB[N*64] | hs[N*64]  (~77 MB)
  float* ws = (float*)d_ws;
  float* dinv = ws;
  float* bufA = ws + (((size_t)N + 255) & ~(size_t)255);
  float* bufB = bufA + (size_t)N * HID;
  float* hs   = bufB + (size_t)N * HID;

  const int NH = N * HID;
  const int gN   = (N + 255) / 256;
  const int gE   = (E + 255) / 256;
  const int gF   = (NH / 4 + 255) / 256;                    // float4 fill
  const int gSc  = (int)(((long long)E * 16 + 255) / 256);  // 16 thr/edge
  const int gNR  = (int)(((long long)N * 16 + 255) / 256);
  const int gGemm = ((N + 15) / 16 + 7) / 8;  // 16 rows/wave, 8 waves/block
  const int gMlp  = ((E + 15) / 16 + 3) / 4;  // 16 edges/wave, 4 waves/block

  // D^{-1/2}  (deg includes self loop via +1 in rsqrt)
  fill_zero_kernel<<<gN, 256, 0, stream>>>(dinv, N);
  deg_kernel<<<gE, 256, 0, stream>>>(dst, dinv, E);
  dinv_kernel<<<gN, 256, 0, stream>>>(dinv, N);

  // Layer 1: hs = (x@W1)*dinv ; acc += hs[src] ; h1 = relu(dinv*(acc+hs)+b1)
  gcn_gemm_scale_kernel<F_IN><<<gGemm, 256, 0, stream>>>(x, W1, dinv, hs, N);
  fill_zero4_kernel<<<gF, 256, 0, stream>>>((float4*)bufA, NH / 4);
  scatter_kernel<<<gSc, 256, 0, stream>>>(hs, src, dst, bufA, E);
  norm_relu_kernel<<<gNR, 256, 0, stream>>>(hs, dinv, b1, bufA, N);

  // Layer 2
  gcn_gemm_scale_kernel<HID><<<gGemm, 256, 0, stream>>>(bufA, W2, dinv, hs, N);
  fill_zero4_kernel<<<gF, 256, 0, stream>>>((float4*)bufB, NH / 4);
  scatter_kernel<<<gSc, 256, 0, stream>>>(hs, src, dst, bufB, E);
  norm_relu_kernel<<<gNR, 256, 0, stream>>>(hs, dinv, b2, bufB, N);

  // Layer 3
  gcn_gemm_scale_kernel<HID><<<gGemm, 256, 0, stream>>>(bufB, W3, dinv, hs, N);
  fill_zero4_kernel<<<gF, 256, 0, stream>>>((float4*)bufA, NH / 4);
  scatter_kernel<<<gSc, 256, 0, stream>>>(hs, src, dst, bufA, E);
  norm_relu_kernel<<<gNR, 256, 0, stream>>>(hs, dinv, b3, bufA, N);

  // Edge classifier MLP
  edge_mlp_kernel<<<gMlp, 128, 0, stream>>>(bufA, src, dst, ea, Wm1, bm1, Wm2,
                                            bm2, Wm3, bm3, out, E);
}